// AdaFace_43542378447384
// MI455X (gfx1250) — compile-verified
//
#include <hip/hip_runtime.h>
#include <math.h>

// ---- problem constants (match reference) ----
#define B_SZ      1024
#define C_SZ      100000
#define SCALE_F   64.0f
#define MARGIN_F  0.4f
#define H_F       0.333f
#define EPS_F     1e-3f
#define PI_F      3.14159265358979323846f

#define VEC_PER_ROW (C_SZ / 4)   // 25000 float4 per row (C divisible by 4)
#define TPB         256
#define UNROLL      4

typedef float f32x4 __attribute__((ext_vector_type(4)));

// ---------------------------------------------------------------------------
// Kernel A: batch stats over norms (wave32 shuffle reduction) + per-row target
// logit adjustment.  Writes SCALE * new_target into row_target[B].
// One block of 1024 threads = 32 wave32 waves.
// ---------------------------------------------------------------------------
__global__ __launch_bounds__(B_SZ)
void adaface_prepass(const float* __restrict__ logits,
                     const float* __restrict__ norms,
                     const int*   __restrict__ labels,
                     float*       __restrict__ row_target)
{
    __shared__ float s_sum[32];
    __shared__ float s_sq[32];
    __shared__ float s_mean, s_invstd_h;   // H / (std + eps)

    const int t    = threadIdx.x;
    const int lane = t & 31;               // wave32 lane
    const int wave = t >> 5;

    // safe_norms = clip(norms, 1e-3, 100)
    float n = norms[t];
    n = fminf(fmaxf(n, 1e-3f), 100.0f);

    // wave32 tree reduction of sum and sum-of-squares
    float s = n;
    float q = n * n;
    #pragma unroll
    for (int off = 16; off > 0; off >>= 1) {
        s += __shfl_xor(s, off, 32);
        q += __shfl_xor(q, off, 32);
    }
    if (lane == 0) { s_sum[wave] = s; s_sq[wave] = q; }
    __syncthreads();

    if (wave == 0) {
        float ss = s_sum[lane];
        float qq = s_sq[lane];
        #pragma unroll
        for (int off = 16; off > 0; off >>= 1) {
            ss += __shfl_xor(ss, off, 32);
            qq += __shfl_xor(qq, off, 32);
        }
        if (lane == 0) {
            const float mean = ss * (1.0f / (float)B_SZ);
            // unbiased (ddof=1) variance: (sum(x^2) - sum(x)*mean) / (B-1)
            float var = (qq - ss * mean) * (1.0f / (float)(B_SZ - 1));
            var = fmaxf(var, 0.0f);
            s_mean     = mean;
            s_invstd_h = H_F / (sqrtf(var) + EPS_F);
        }
    }
    __syncthreads();

    // per-row margin scaler and target-logit rewrite
    float ms = (n - s_mean) * s_invstd_h;
    ms = fminf(fmaxf(ms, -1.0f), 1.0f);

    const float g_ang = -MARGIN_F * ms;            // additive angular margin
    const float g_add =  MARGIN_F + MARGIN_F * ms; // additive cosine margin

    const int   lbl = labels[t];
    const float tl  = logits[(size_t)t * (size_t)C_SZ + (size_t)lbl];

    float theta = acosf(tl) + g_ang;
    theta = fminf(fmaxf(theta, EPS_F), PI_F - EPS_F);
    const float new_target = cosf(theta) - g_add;

    row_target[t] = new_target * SCALE_F;
}

// ---------------------------------------------------------------------------
// Kernel B: streaming scale of the 1024x100000 matrix.
//   out = logits * 64, with out[row, labels[row]] = row_target[row].
// b128 (float4) accesses with non-temporal TH hints: 800 MB streams exactly
// once, keep it out of the 192 MB L2.  Grid: (ceil(25000/1024), 1024 rows).
// ---------------------------------------------------------------------------
__global__ __launch_bounds__(TPB)
void adaface_scale(const float* __restrict__ logits,
                   const int*   __restrict__ labels,
                   const float* __restrict__ row_target,
                   float*       __restrict__ out)
{
    const int row = blockIdx.y;

    // block-uniform scalars
    const int   lbl     = labels[row];
    const float tval    = row_target[row];
    const int   lblVec  = lbl >> 2;   // which float4 holds the target column
    const int   lblComp = lbl & 3;    // which component within it

    const f32x4* __restrict__ src = (const f32x4*)(logits + (size_t)row * (size_t)C_SZ);
    f32x4*       __restrict__ dst = (f32x4*)      (out    + (size_t)row * (size_t)C_SZ);

    const int base = blockIdx.x * (TPB * UNROLL) + threadIdx.x;

    int   idx[UNROLL];
    f32x4 v[UNROLL];

    // issue all loads first (4 x b128 NT loads in flight per thread)
    #pragma unroll
    for (int u = 0; u < UNROLL; ++u) {
        idx[u] = base + u * TPB;
        if (idx[u] < VEC_PER_ROW)
            v[u] = __builtin_nontemporal_load(&src[idx[u]]);
    }

    #pragma unroll
    for (int u = 0; u < UNROLL; ++u) {
        if (idx[u] < VEC_PER_ROW) {
            f32x4 r = v[u] * SCALE_F;
            if (idx[u] == lblVec) {         // at most one thread per row takes this
                if      (lblComp == 0) r.x = tval;
                else if (lblComp == 1) r.y = tval;
                else if (lblComp == 2) r.z = tval;
                else                   r.w = tval;
            }
            __builtin_nontemporal_store(r, &dst[idx[u]]);
        }
    }
}

// ---------------------------------------------------------------------------
extern "C" void kernel_launch(void* const* d_in, const int* in_sizes, int n_in,
                              void* d_out, int out_size, void* d_ws, size_t ws_size,
                              hipStream_t stream)
{
    (void)in_sizes; (void)n_in; (void)out_size; (void)ws_size;

    const float* logits = (const float*)d_in[0];  // [B, C] fp32
    const float* norms  = (const float*)d_in[1];  // [B, 1] fp32
    const int*   labels = (const int*)  d_in[2];  // [B]    int32
    float*       out    = (float*)d_out;          // [B, C] fp32
    float*       row_t  = (float*)d_ws;           // [B] scratch: SCALE*new_target

    // 1) per-row adjusted target logits (tiny: one block, 32 waves)
    adaface_prepass<<<dim3(1), dim3(B_SZ), 0, stream>>>(logits, norms, labels, row_t);

    // 2) streaming scale + scatter fixup (memory-bound, ~800 MB @ 23.3 TB/s HBM)
    dim3 grid((VEC_PER_ROW + TPB * UNROLL - 1) / (TPB * UNROLL), B_SZ);
    adaface_scale<<<grid, dim3(TPB), 0, stream>>>(logits, labels, row_t, out);
}